// CA_HGER_Model_84782654423762
// MI455X (gfx1250) — compile-verified
//
#include <hip/hip_runtime.h>
#include <hip/hip_bf16.h>

typedef float v2f __attribute__((ext_vector_type(2)));
typedef float v8f __attribute__((ext_vector_type(8)));
typedef unsigned int v4u __attribute__((ext_vector_type(4)));
typedef int v8i __attribute__((ext_vector_type(8)));
typedef int v4i __attribute__((ext_vector_type(4)));

#define N_NODES   100000
#define N_EDGES   400000
#define EMB       128
#define HID       256
#define BATCH     1024
#define BN_EPS    1e-5f

#if defined(__has_builtin)
#if __has_builtin(__builtin_amdgcn_tensor_load_to_lds)
#define USE_TDM 1
#else
#define USE_TDM 0
#endif
#else
#define USE_TDM 0
#endif

// ---------------------------------------------------------------------------
// Generic fp32 WMMA GEMM: C[M,N] = A[M,K] @ B[K,N] (+bias) (+relu)
// A,B,C row-major. K multiple of 32, N multiple of 128 per grid.y slab.
// Block = 256 threads = 8 waves; workgroup tile = 128 rows x 128 cols;
// wave tile = 16 rows x 128 cols (8 accumulators), K stepped by 4 via
// v_wmma_f32_16x16x4_f32.
// Per 32-wide K chunk:
//   A panel (128x32) -> LDS via TDM tensor_load_to_lds with pad_enable
//     (2 DWORD pad per 32-DWORD row => stride 34 => banks (2r+k)%64, conflict-free)
//   B chunk staged transposed (n-major, k-contiguous, stride 34) so each
//     B fragment is one aligned ds_load_b64 (b.x=B[k][n], b.y=B[k+1][n]).
// ---------------------------------------------------------------------------
__global__ __launch_bounds__(256)
void wmma_gemm_f32(const float* __restrict__ A, const float* __restrict__ B,
                   float* __restrict__ C, const float* __restrict__ bias,
                   int M, int N, int K, int relu)
{
    __shared__ float As[128 * 34];   // 17 KB, [row][k'] stride 34 (TDM-padded)
    __shared__ float Bs[128 * 34];   // 17 KB, transposed chunk [n][k'] stride 34

    const int wave = threadIdx.x >> 5;
    const int lane = threadIdx.x & 31;
    const int row_base = blockIdx.x * 128 + wave * 16;
    const int col_base = blockIdx.y * 128;
    const int row0 = blockIdx.x * 128;

    v8f acc[8];
#pragma unroll
    for (int t = 0; t < 8; ++t) acc[t] = (v8f){0.f,0.f,0.f,0.f,0.f,0.f,0.f,0.f};

    // A fragment (16x4 layout): lanes 0-15 hold K=k,k+1; lanes 16-31 hold K=k+2,k+3
    const int k_half = (lane >> 4) << 1;          // 0 or 2
    const int n16 = lane & 15;
    const int rloc = (wave << 4) + n16;           // local A row 0..127

    for (int kc = 0; kc < K; kc += 32) {
        __syncthreads();

#if USE_TDM
        if (wave == 0) {
            // Tensor DMA: 2D tile (tile_dim0=32 elems, tile_dim1=128 rows),
            // stride0 = K, dims relative to tile start for OOB zero-fill.
            const unsigned lds_base = (unsigned)(size_t)(&As[0]);
            const unsigned long long ga =
                (unsigned long long)(size_t)(A + (size_t)row0 * K + kc);
            const unsigned tdim0 = (unsigned)(K - kc);      // remaining row length
            const unsigned tdim1 = (unsigned)(M - row0);    // remaining rows
            const unsigned stride0 = (unsigned)K;

            v4u g0;
            g0.x = 1u;                                       // count=1, user mode
            g0.y = lds_base;                                 // lds_addr [63:32]
            g0.z = (unsigned)ga;                             // global_addr lo
            g0.w = (unsigned)((ga >> 32) & 0x1FFFFFFull)     // global_addr [56:32]
                 | (2u << 30);                               // type=2 (image)

            v8i g1;
            g1[0] = (2 << 16)          // data_size = 4 bytes
                  | (1 << 20)          // pad_enable
                  | (4 << 22)          // pad_interval: every 32 DWORDs
                  | (1 << 25);         // pad_amount: 2 DWORDs
            g1[1] = (int)((tdim0 & 0xffffu) << 16);                      // tensor_dim0 lo16
            g1[2] = (int)(((tdim0 >> 16) & 0xffffu) |
                          ((tdim1 & 0xffffu) << 16));                    // dim0 hi / dim1 lo
            g1[3] = (int)(((tdim1 >> 16) & 0xffffu) | (32u << 16));      // dim1 hi / tile_dim0=32
            g1[4] = 128;                                                 // tile_dim1=128, tile_dim2=0
            g1[5] = (int)stride0;                                        // tensor_dim0_stride lo32
            g1[6] = 0;
            g1[7] = 0;

            v4i z4 = {0, 0, 0, 0};
            v8i z8 = {0, 0, 0, 0, 0, 0, 0, 0};
            __builtin_amdgcn_tensor_load_to_lds(g0, g1, z4, z4, z8, 0);
        }
#else
        // fallback: manual A staging (row-major, stride 34)
        for (int i = threadIdx.x; i < 128 * 32; i += 256) {
            int rr = i >> 5, kk = i & 31;
            int gr = row0 + rr; if (gr >= M) gr = M - 1;
            As[rr * 34 + kk] = A[(size_t)gr * K + kc + kk];
        }
#endif

        // stage B[kc:kc+32, col_base:col_base+128] transposed into LDS
        // global read coalesced over nn; LDS write banks (2*nn+kk)%64 conflict-free
        for (int i = threadIdx.x; i < 32 * 128; i += 256) {
            int kk = i >> 7, nn = i & 127;
            Bs[nn * 34 + kk] = B[(size_t)(kc + kk) * N + col_base + nn];
        }

#if USE_TDM
        if (wave == 0) __builtin_amdgcn_s_wait_tensorcnt((short)0);
#endif
        __syncthreads();

#pragma unroll
        for (int ks = 0; ks < 32; ks += 4) {
            const int bofs = ks + k_half;          // even -> 8B aligned ds_load_b64
            v2f a = *(const v2f*)(&As[rloc * 34 + bofs]);
#pragma unroll
            for (int t = 0; t < 8; ++t) {
                v2f b = *(const v2f*)(&Bs[(t * 16 + n16) * 34 + bofs]);
                acc[t] = __builtin_amdgcn_wmma_f32_16x16x4_f32(
                    false, a, false, b, (short)0, acc[t], false, false);
            }
        }
    }

    // store: C/D layout — VGPR i: lanes 0-15 -> M=i, lanes 16-31 -> M=8+i; N=lane%16
    const int rlo = row_base + ((lane >> 4) << 3);
#pragma unroll
    for (int t = 0; t < 8; ++t) {
        const int col = col_base + t * 16 + n16;
        const float bv = bias ? bias[col] : 0.f;
#pragma unroll
        for (int i = 0; i < 8; ++i) {
            const int row = rlo + i;
            if (row < M) {
                float v = acc[t][i] + bv;
                if (relu) v = fmaxf(v, 0.f);
                C[(size_t)row * N + col] = v;
            }
        }
    }
}

// ---------------------------------------------------------------------------
// Edge scatter for one relation: acc[dst] += ht[src] * w   (wave per edge,
// grid-stride; prefetch next edge's source row to hide gather latency)
// ---------------------------------------------------------------------------
__global__ __launch_bounds__(256)
void edge_scatter(const float* __restrict__ ht, float* __restrict__ acc,
                  const long long* __restrict__ src, const long long* __restrict__ dst,
                  const long long* __restrict__ etype, const float* __restrict__ ew,
                  int rel, int n_edges)
{
    const int wave = threadIdx.x >> 5;
    const int lane = threadIdx.x & 31;
    const int stride = gridDim.x * 8;

    for (int e = blockIdx.x * 8 + wave; e < n_edges; e += stride) {
        const int en = e + stride;
        if (en < n_edges) {
            // speculative warm of next source row (global_prefetch_b8)
            __builtin_prefetch(ht + (size_t)src[en] * EMB + lane * 4, 0, 1);
        }
        if ((int)etype[e] != rel) continue;

        const int s = (int)src[e];
        const int d = (int)dst[e];
        const float w = (rel == 1) ? (ew[e] + 1.0f) * 0.5f : 1.0f;

        float4 v = ((const float4*)(ht + (size_t)s * EMB))[lane];
        float* drow = acc + (size_t)d * EMB + lane * 4;
        atomicAdd(drow + 0, v.x * w);
        atomicAdd(drow + 1, v.y * w);
        atomicAdd(drow + 2, v.z * w);
        atomicAdd(drow + 3, v.w * w);
    }
}

// ---------------------------------------------------------------------------
// BatchNorm: column sums / sums-of-squares (blockDim.x == feats)
// ---------------------------------------------------------------------------
__global__ void bn_stats(const float* __restrict__ x, float* __restrict__ sums,
                         float* __restrict__ sqs, int rows, int feats, int chunk)
{
    const int col = threadIdx.x;
    int r0 = blockIdx.x * chunk;
    int r1 = r0 + chunk; if (r1 > rows) r1 = rows;
    float s = 0.f, q = 0.f;
    for (int r = r0; r < r1; ++r) {
        float v = x[(size_t)r * feats + col];
        s += v; q += v * v;
    }
    atomicAdd(&sums[col], s);
    atomicAdd(&sqs[col], q);
}

__global__ void bn_apply_relu(const float* __restrict__ x, float* __restrict__ y,
                              const float* __restrict__ sums, const float* __restrict__ sqs,
                              const float* __restrict__ gamma, const float* __restrict__ beta,
                              int rows, int feats)
{
    const int idx = blockIdx.x * blockDim.x + threadIdx.x;
    if (idx >= rows * feats) return;
    const int col = idx % feats;
    const float inv_n = 1.0f / (float)rows;
    const float mu = sums[col] * inv_n;
    const float var = sqs[col] * inv_n - mu * mu;
    float v = (x[idx] - mu) * rsqrtf(var + BN_EPS) * gamma[col] + beta[col];
    y[idx] = fmaxf(v, 0.f);
}

// ---------------------------------------------------------------------------
// Gather claim rows; final tiny linear (128 -> 2)
// ---------------------------------------------------------------------------
__global__ void gather_rows(const float* __restrict__ h, const long long* __restrict__ idx,
                            float* __restrict__ out)
{
    const long long n = idx[blockIdx.x];
    out[(size_t)blockIdx.x * EMB + threadIdx.x] = h[(size_t)n * EMB + threadIdx.x];
}

__global__ void final_linear(const float* __restrict__ x, const float* __restrict__ W3,
                             const float* __restrict__ b3, float* __restrict__ out)
{
    const int idx = blockIdx.x * blockDim.x + threadIdx.x;
    if (idx >= BATCH * 2) return;
    const int b = idx >> 1, j = idx & 1;
    float s = b3[j];
    const float* xr = x + (size_t)b * EMB;
#pragma unroll 8
    for (int k = 0; k < EMB; ++k) s += xr[k] * W3[k * 2 + j];
    out[idx] = s;
}

// ---------------------------------------------------------------------------
extern "C" void kernel_launch(void* const* d_in, const int* in_sizes, int n_in,
                              void* d_out, int out_size, void* d_ws, size_t ws_size,
                              hipStream_t stream) {
    const float*     nodef = (const float*)d_in[0];
    const long long* eidx  = (const long long*)d_in[1];
    const long long* etyp  = (const long long*)d_in[2];
    const float*     ew    = (const float*)d_in[3];
    const long long* claim = (const long long*)d_in[4];
    const float*     gnnW  = (const float*)d_in[5];
    const float*     ggam  = (const float*)d_in[6];
    const float*     gbet  = (const float*)d_in[7];
    const float*     W1    = (const float*)d_in[8];
    const float*     b1    = (const float*)d_in[9];
    const float*     gamc  = (const float*)d_in[10];
    const float*     betc  = (const float*)d_in[11];
    const float*     W2    = (const float*)d_in[12];
    const float*     b2    = (const float*)d_in[13];
    const float*     W3    = (const float*)d_in[14];
    const float*     b3    = (const float*)d_in[15];
    float* out = (float*)d_out;

    const long long* src = eidx;
    const long long* dst = eidx + N_EDGES;

    // workspace layout (floats)
    const size_t NF = (size_t)N_NODES * EMB;       // 12.8M
    float* ws   = (float*)d_ws;
    float* hA   = ws;                               // layer output features
    float* acc  = hA  + NF;                         // aggregation accumulator
    float* ht   = acc + NF;                         // per-relation transformed feats
    float* cbuf = ht  + NF;                         // (1024,128)
    float* x1   = cbuf + (size_t)BATCH * EMB;       // (1024,256)
    float* x2   = x1   + (size_t)BATCH * HID;       // (1024,128)
    float* sums = x2   + (size_t)BATCH * EMB;       // 256
    float* sqs  = sums + 256;                       // 256

    const int gemm_gx_nodes = (N_NODES + 127) / 128;   // 782
    const int edge_blocks   = 12500;                   // 4 edges per wave

    for (int l = 0; l < 2; ++l) {
        const float* hcur = (l == 0) ? nodef : hA;

        // residual init: acc = hcur
        (void)hipMemcpyAsync(acc, hcur, NF * sizeof(float), hipMemcpyDeviceToDevice, stream);

        for (int r = 0; r < 3; ++r) {
            const float* Wr = gnnW + ((size_t)(l * 3 + r)) * EMB * EMB;
            // ht = hcur @ W[l,r]
            wmma_gemm_f32<<<dim3(gemm_gx_nodes, 1), 256, 0, stream>>>(
                hcur, Wr, ht, nullptr, N_NODES, EMB, EMB, 0);
            // acc[dst] += ht[src] * w  (relation-masked)
            edge_scatter<<<edge_blocks, 256, 0, stream>>>(
                ht, acc, src, dst, etyp, ew, r, N_EDGES);
        }

        // BN (batch stats) + ReLU -> hA
        (void)hipMemsetAsync(sums, 0, 512 * sizeof(float), stream);
        bn_stats<<<(N_NODES + 511) / 512, EMB, 0, stream>>>(acc, sums, sqs, N_NODES, EMB, 512);
        bn_apply_relu<<<(int)(NF / 256), 256, 0, stream>>>(
            acc, hA, sums, sqs, ggam + l * EMB, gbet + l * EMB, N_NODES, EMB);
    }

    // classifier head
    gather_rows<<<BATCH, EMB, 0, stream>>>(hA, claim, cbuf);

    // x1 = c @ W1 + b1  (1024x128 @ 128x256)
    wmma_gemm_f32<<<dim3(BATCH / 128, HID / 128), 256, 0, stream>>>(
        cbuf, W1, x1, b1, BATCH, HID, EMB, 0);

    (void)hipMemsetAsync(sums, 0, 512 * sizeof(float), stream);
    bn_stats<<<BATCH / 256, HID, 0, stream>>>(x1, sums, sqs, BATCH, HID, 256);
    bn_apply_relu<<<(BATCH * HID) / 256, 256, 0, stream>>>(
        x1, x1, sums, sqs, gamc, betc, BATCH, HID);

    // x2 = relu(x1 @ W2 + b2)  (1024x256 @ 256x128)
    wmma_gemm_f32<<<dim3(BATCH / 128, 1), 256, 0, stream>>>(
        x1, W2, x2, b2, BATCH, EMB, HID, 1);

    // logits = x2 @ W3 + b3
    final_linear<<<(BATCH * 2 + 255) / 256, 256, 0, stream>>>(x2, W3, b3, out);
}